// BahdanauAttention_24833500906223
// MI455X (gfx1250) — compile-verified
//
#include <hip/hip_runtime.h>
#include <hip/hip_bf16.h>

// ---------------------------------------------------------------------------
// Bahdanau additive attention for MI455X (gfx1250, wave32, WMMA).
// B=4, LQ=LK=512, H=256.
// ---------------------------------------------------------------------------

typedef __attribute__((ext_vector_type(2))) float v2f;
typedef __attribute__((ext_vector_type(8))) float v8f;

#define NB 4
#define LQ 512
#define LK 512
#define HD 256
#define ROWS (NB * LQ)   // 2048 flattened (b, q) rows

// ---- CDNA5 hardware transcendentals ---------------------------------------
__device__ __forceinline__ float fast_tanh(float x) {
#if __has_builtin(__builtin_amdgcn_tanhf)
  return __builtin_amdgcn_tanhf(x);          // v_tanh_f32 (TRANS unit)
#else
  float y;
  asm volatile("v_tanh_f32 %0, %1\n\tv_nop" : "=v"(y) : "v"(x));
  return y;
#endif
}

__device__ __forceinline__ float fast_exp(float x) {
#if __has_builtin(__builtin_amdgcn_exp2f)
  return __builtin_amdgcn_exp2f(x * 1.44269504088896340736f);  // v_exp_f32
#else
  return __expf(x);
#endif
}

// ---------------------------------------------------------------------------
// K1: projection GEMM  Y(M,256) = X(M,256) * W(256,256)^T + bias
// One wave per 16x16 output tile, V_WMMA_F32_16X16X4_F32, K stepped by 4.
// ---------------------------------------------------------------------------
__global__ __launch_bounds__(32) void proj_wmma_kernel(
    const float* __restrict__ X,     // (ROWS, 256)
    const float* __restrict__ W,     // (256, 256) row-major (o, h)
    const float* __restrict__ bias,  // (256)
    float* __restrict__ Y)           // (ROWS, 256)
{
  const int lane  = threadIdx.x;
  const int nbase = blockIdx.x * 16;
  const int mbase = blockIdx.y * 16;
  const int ml    = lane & 15;
  const int kph   = (lane >> 4) << 1;  // 0 or 2

  const float* __restrict__ arow = X + (size_t)(mbase + ml) * HD;
  const float* __restrict__ brow = W + (size_t)(nbase + ml) * HD;

  v8f c = {};
  #pragma unroll 4
  for (int kk = 0; kk < HD; kk += 4) {
    v2f a = *(const v2f*)(arow + kk + kph);
    v2f b = *(const v2f*)(brow + kk + kph);
    c = __builtin_amdgcn_wmma_f32_16x16x4_f32(false, a, false, b,
                                              (short)0, c, false, false);
  }

  const int col = nbase + ml;
  const int mo  = mbase + ((lane >> 4) << 3);
  const float bv = bias[col];
  #pragma unroll
  for (int v = 0; v < 8; ++v)
    Y[(size_t)(mo + v) * HD + col] = c[v] + bv;
}

// ---------------------------------------------------------------------------
// K2: scores[b,q,k] = sum_h We[h] * tanh(qp[b,q,h] + kp[b,k,h]) + be
//
// Macro tile 32q x 64k per 256-thread block; micro tile 2q x 4k per thread.
// LDS layouts chosen for zero bank conflicts:
//   sQ : row-padded to 257 floats  -> the 2 q-rows read by each wave-half
//        sit in distinct banks; reads are broadcasts.
//   sKT: k-proj tile stored TRANSPOSED [h][k], row = 64 floats -> per-h
//        b128 read covers banks 0..63 exactly once.
// q-tile + We staged with CDNA5 global_load_async_to_lds_b128 (ASYNCcnt),
// per-lane LDS destinations write straight into the padded layout.
// ---------------------------------------------------------------------------
#define TQ 32
#define TK 64
#define QPAD 257

__global__ __launch_bounds__(256) void score_kernel(
    const float* __restrict__ QP,   // (ROWS, 256)
    const float* __restrict__ KP,   // (ROWS, 256)
    const float* __restrict__ We,   // (256)
    const float* __restrict__ be,   // (1)
    float* __restrict__ S)          // (NB, LQ, LK)  raw scores
{
  __shared__ __align__(16) float sQ[TQ * QPAD];   // ~32.1 KB (padded rows)
  __shared__ __align__(16) float sKT[HD * TK];    //  64 KB (transposed)
  __shared__ __align__(16) float sW[HD];          //   1 KB

  const int b     = blockIdx.z;
  const int qbase = blockIdx.y * TQ;
  const int kbase = blockIdx.x * TK;
  const int tid   = threadIdx.x;

  // --- async-copy q tile (contiguous global -> padded LDS rows) -----------
  {
    const float4* gq = (const float4*)(QP + (size_t)(b * LQ + qbase) * HD);
    #pragma unroll
    for (int i = 0; i < (TQ * HD / 4) / 256; ++i) {   // 8 float4 per thread
      const int f4 = tid + i * 256;
      const int r  = f4 >> 6;           // 64 float4 per q-row
      const int c4 = f4 & 63;
      unsigned lds = (unsigned)(size_t)(sQ + r * QPAD + c4 * 4);
      asm volatile("global_load_async_to_lds_b128 %0, %1, off"
                   :: "v"(lds), "v"(gq + f4) : "memory");
    }
    if (tid < HD / 4) {
      unsigned lds = (unsigned)(size_t)(sW + tid * 4);
      asm volatile("global_load_async_to_lds_b128 %0, %1, off"
                   :: "v"(lds), "v"(((const float4*)We) + tid) : "memory");
    }
  }

  // --- manual transpose-staging of k tile into sKT[h][k] -------------------
  {
    const int k  = tid & 63;
    const int hc = (tid >> 6) * 64;     // each thread owns 64 h-values of one k
    const float4* gk =
        (const float4*)(KP + (size_t)(b * LK + kbase + k) * HD + hc);
    #pragma unroll 4
    for (int j = 0; j < 16; ++j) {
      const float4 v = gk[j];
      const int h = hc + j * 4;
      sKT[(h + 0) * TK + k] = v.x;      // store banks == k : conflict-free
      sKT[(h + 1) * TK + k] = v.y;
      sKT[(h + 2) * TK + k] = v.z;
      sKT[(h + 3) * TK + k] = v.w;
    }
  }

  asm volatile("s_wait_asynccnt 0x0" ::: "memory");
  __syncthreads();

  // --- TRANS-bound main loop: 8 tanh per h-iteration per thread ------------
  const int tqr = tid >> 4;             // 0..15  -> q rows 2*tqr, 2*tqr+1
  const int tkc = (tid & 15) << 2;      // 0,4,...,60 -> 4 k columns

  const float* __restrict__ q0 = sQ + (tqr * 2) * QPAD;
  const float* __restrict__ q1 = q0 + QPAD;
  const float* __restrict__ kt = sKT + tkc;

  float a00 = 0.f, a01 = 0.f, a02 = 0.f, a03 = 0.f;
  float a10 = 0.f, a11 = 0.f, a12 = 0.f, a13 = 0.f;

  #pragma unroll 4
  for (int h = 0; h < HD; ++h) {
    const float  w  = sW[h];
    const float  x0 = q0[h];
    const float  x1 = q1[h];
    const float4 kv = *(const float4*)(kt + h * TK);
    a00 += w * fast_tanh(x0 + kv.x);
    a01 += w * fast_tanh(x0 + kv.y);
    a02 += w * fast_tanh(x0 + kv.z);
    a03 += w * fast_tanh(x0 + kv.w);
    a10 += w * fast_tanh(x1 + kv.x);
    a11 += w * fast_tanh(x1 + kv.y);
    a12 += w * fast_tanh(x1 + kv.z);
    a13 += w * fast_tanh(x1 + kv.w);
  }

  const float bev = be[0];
  float* out = S + (size_t)(b * LQ + qbase + tqr * 2) * LK + kbase + tkc;
  *(float4*)out        = make_float4(a00 + bev, a01 + bev, a02 + bev, a03 + bev);
  *(float4*)(out + LK) = make_float4(a10 + bev, a11 + bev, a12 + bev, a13 + bev);
}

// ---------------------------------------------------------------------------
// K3: in-place softmax over last axis (row length 512). One wave32 per row.
// ---------------------------------------------------------------------------
__global__ __launch_bounds__(256) void softmax_kernel(float* __restrict__ A)
{
  const int row  = blockIdx.x * 8 + (threadIdx.x >> 5);
  const int lane = threadIdx.x & 31;
  float* __restrict__ a = A + (size_t)row * LK;

  float vals[16];
  float m = -3.0e38f;
  #pragma unroll
  for (int i = 0; i < 16; ++i) {
    vals[i] = a[lane + i * 32];
    m = fmaxf(m, vals[i]);
  }
  #pragma unroll
  for (int off = 16; off > 0; off >>= 1)
    m = fmaxf(m, __shfl_xor(m, off, 32));

  float sum = 0.f;
  #pragma unroll
  for (int i = 0; i < 16; ++i) {
    vals[i] = fast_exp(vals[i] - m);
    sum += vals[i];
  }
  #pragma unroll
  for (int off = 16; off > 0; off >>= 1)
    sum += __shfl_xor(sum, off, 32);

  const float inv = 1.0f / sum;
  #pragma unroll
  for (int i = 0; i < 16; ++i)
    a[lane + i * 32] = vals[i] * inv;
}

// ---------------------------------------------------------------------------
// K4: context[b,q,h] = sum_k attn[b,q,k] * values[b,k,h]
// Batched GEMM M=512, N=256, K=512 with V_WMMA_F32_16X16X4_F32.
// ---------------------------------------------------------------------------
__global__ __launch_bounds__(32) void context_wmma_kernel(
    const float* __restrict__ Attn,  // (NB, LQ, LK)
    const float* __restrict__ V,     // (NB, LK, HD)
    float* __restrict__ C)           // (NB, LQ, HD)
{
  const int b     = blockIdx.z;
  const int nbase = blockIdx.x * 16;
  const int mbase = blockIdx.y * 16;
  const int lane  = threadIdx.x;
  const int ml    = lane & 15;
  const int kph   = (lane >> 4) << 1;

  const float* __restrict__ arow = Attn + (size_t)(b * LQ + mbase + ml) * LK;
  const float* __restrict__ vb   = V + (size_t)b * LK * HD + (nbase + ml);

  v8f c = {};
  #pragma unroll 4
  for (int kk = 0; kk < LK; kk += 4) {
    v2f a = *(const v2f*)(arow + kk + kph);
    v2f bf;
    bf.x = vb[(size_t)(kk + kph) * HD];
    bf.y = vb[(size_t)(kk + kph + 1) * HD];
    c = __builtin_amdgcn_wmma_f32_16x16x4_f32(false, a, false, bf,
                                              (short)0, c, false, false);
  }

  const int col = nbase + ml;
  const int mo  = mbase + ((lane >> 4) << 3);
  #pragma unroll
  for (int v = 0; v < 8; ++v)
    C[(size_t)(b * LQ + mo + v) * HD + col] = c[v];
}

// ---------------------------------------------------------------------------
// Host launch. Inputs (all f32): query, values, Wq, bq, Wk, bk, We, be.
// d_out = [context (4*512*256) | attention (4*512*512)] f32.
// Workspace: qp (2 MB) + kp (2 MB). Raw scores go straight into the
// attention region of d_out and are softmaxed in place.
// ---------------------------------------------------------------------------
extern "C" void kernel_launch(void* const* d_in, const int* in_sizes, int n_in,
                              void* d_out, int out_size, void* d_ws, size_t ws_size,
                              hipStream_t stream) {
  const float* query  = (const float*)d_in[0];
  const float* values = (const float*)d_in[1];
  const float* Wq     = (const float*)d_in[2];
  const float* bq     = (const float*)d_in[3];
  const float* Wk     = (const float*)d_in[4];
  const float* bk     = (const float*)d_in[5];
  const float* We     = (const float*)d_in[6];
  const float* be     = (const float*)d_in[7];

  float* qp   = (float*)d_ws;                    // (2048, 256)
  float* kp   = qp + (size_t)ROWS * HD;          // (2048, 256)
  float* ctx  = (float*)d_out;                   // (4, 512, 256)
  float* attn = ctx + (size_t)NB * LQ * HD;      // (4, 512, 512)

  // K1: projections (M=2048, N=256, K=256), one wave per 16x16 tile.
  dim3 pgrid(HD / 16, ROWS / 16);
  proj_wmma_kernel<<<pgrid, 32, 0, stream>>>(query,  Wq, bq, qp);
  proj_wmma_kernel<<<pgrid, 32, 0, stream>>>(values, Wk, bk, kp);

  // K2: tanh-energy scores -> attention region of d_out (raw).
  dim3 sgrid(LK / TK, LQ / TQ, NB);              // (8, 16, 4) = 512 blocks
  score_kernel<<<sgrid, 256, 0, stream>>>(qp, kp, We, be, attn);

  // K3: softmax in place (2048 rows, 8 rows per 256-thread block).
  softmax_kernel<<<ROWS / 8, 256, 0, stream>>>(attn);

  // K4: context = attention @ values.
  dim3 cgrid(HD / 16, LQ / 16, NB);
  context_wmma_kernel<<<cgrid, 32, 0, stream>>>(attn, values, ctx);
}